// SAGEConv_91250875171573
// MI455X (gfx1250) — compile-verified
//
#include <hip/hip_runtime.h>

#define N_NODES 100000
#define N_EDGES 640000
#define D       128
#define ROWTILES 4            // 16-row tiles per wave -> 64 rows per block

typedef __attribute__((ext_vector_type(2))) float v2f;
typedef __attribute__((ext_vector_type(8))) float v8f;

// ---------------- Kernel 1: zero workspace (sum[N,128] ++ deg[N]) ----------
__global__ __launch_bounds__(256) void sage_zero_kernel(float4* __restrict__ ws, int n4) {
    int i = blockIdx.x * 256 + threadIdx.x;
    if (i < n4) ws[i] = make_float4(0.f, 0.f, 0.f, 0.f);
}

// ---------------- Kernel 2: edge scatter-add (one wave per edge) -----------
// lane L handles channels [4L, 4L+4). x rows are 512B-aligned so float4 ok.
// x (51 MB) and sum (51 MB) both fit in the 192 MB L2 -> L2-atomic bound.
__global__ __launch_bounds__(256) void sage_scatter_kernel(
    const float* __restrict__ x, const int* __restrict__ esrc,
    const int* __restrict__ edst, float* __restrict__ sum,
    float* __restrict__ deg) {
    int t = blockIdx.x * 256 + threadIdx.x;
    int e = t >> 5;
    if (e >= N_EDGES) return;
    int lane = t & 31;
    int s = esrc[e];
    int d = edst[e];
    const float4 v = *reinterpret_cast<const float4*>(x + (size_t)s * D + lane * 4);
    float* o = sum + (size_t)d * D + lane * 4;
    atomicAdd(o + 0, v.x);
    atomicAdd(o + 1, v.y);
    atomicAdd(o + 2, v.z);
    atomicAdd(o + 3, v.w);
    if (lane == 0) atomicAdd(deg + d, 1.0f);
}

// ---------------- Kernel 3: deg -> 1/max(deg,1) in place -------------------
__global__ __launch_bounds__(256) void sage_invdeg_kernel(float* __restrict__ deg) {
    int i = blockIdx.x * 256 + threadIdx.x;
    if (i < N_NODES) deg[i] = 1.0f / fmaxf(deg[i], 1.0f);
}

// ---------------- Kernel 4: out = (sum*inv) @ W + x @ Wr + bias ------------
// 256 threads = 8 waves; wave w owns output columns [16w, 16w+16).
// Each wave computes ROWTILES=4 independent 16x16 tiles (64 rows / block):
//  - B fragments (W, Wr) loaded once per K-step, reused by 4 A-tiles (4x less
//    weight traffic through L1/L2)
//  - 4 independent accumulator chains -> WMMA latency hiding on the XDL pipe
//
// Fragment layouts (ISA 7.12.2, f32 16x16x4):
//  A: lane<16 -> row=lane,   K={k,k+1}; lane>=16 -> row=lane-16, K={k+2,k+3}
//  B: mirrored, N = lane&15, same K split
//  C/D: VGPR i -> M = i + 8*(lane>=16), N = lane&15
// Rows are clamped (not masked) so EXEC stays all-1s for WMMA; stores masked.
__global__ __launch_bounds__(256) void sage_gemm_kernel(
    const float* __restrict__ sum, const float* __restrict__ invd,
    const float* __restrict__ x, const float* __restrict__ W,
    const float* __restrict__ Wr, const float* __restrict__ bias,
    float* __restrict__ out) {
    const int lane = threadIdx.x & 31;
    const int wave = threadIdx.x >> 5;       // column tile 0..7
    const int n0   = wave * 16;
    const int l16  = lane & 15;
    const int half = lane >> 4;              // 0: K pair {0,1}; 1: K pair {2,3}
    const int row0 = blockIdx.x * (16 * ROWTILES);

    const float* srow[ROWTILES];
    const float* xrow[ROWTILES];
    float        id[ROWTILES];
#pragma unroll
    for (int t = 0; t < ROWTILES; ++t) {
        int r = row0 + 16 * t + l16;
        if (r > N_NODES - 1) r = N_NODES - 1;   // clamp: keep all lanes active
        srow[t] = sum + (size_t)r * D;
        xrow[t] = x   + (size_t)r * D;
        id[t]   = invd[r];
    }

    v8f acc[ROWTILES];
#pragma unroll
    for (int t = 0; t < ROWTILES; ++t) acc[t] = {};

#pragma unroll 2
    for (int k = 0; k < D; k += 4) {
        const int ka = k + 2 * half;
        // B fragments: shared by all 4 row tiles
        const float* wp = W  + (size_t)ka * D + n0 + l16;
        const float* rp = Wr + (size_t)ka * D + n0 + l16;
        v2f bW, bR;
        bW.x = wp[0]; bW.y = wp[D];
        bR.x = rp[0]; bR.y = rp[D];
#pragma unroll
        for (int t = 0; t < ROWTILES; ++t) {
            v2f aM, aX;
            aM.x = srow[t][ka]     * id[t];     // fused mean divide
            aM.y = srow[t][ka + 1] * id[t];
            aX.x = xrow[t][ka];
            aX.y = xrow[t][ka + 1];
            acc[t] = __builtin_amdgcn_wmma_f32_16x16x4_f32(
                false, aM, false, bW, (short)0, acc[t], false, false);
            acc[t] = __builtin_amdgcn_wmma_f32_16x16x4_f32(
                false, aX, false, bR, (short)0, acc[t], false, false);
        }
    }

    const float b = bias[n0 + l16];
#pragma unroll
    for (int t = 0; t < ROWTILES; ++t) {
        const int m0 = row0 + 16 * t + half * 8;
#pragma unroll
        for (int i = 0; i < 8; ++i) {
            const int m = m0 + i;
            if (m < N_NODES)
                out[(size_t)m * D + n0 + l16] = acc[t][i] + b;
        }
    }
}

// ---------------------------------------------------------------------------
extern "C" void kernel_launch(void* const* d_in, const int* in_sizes, int n_in,
                              void* d_out, int out_size, void* d_ws, size_t ws_size,
                              hipStream_t stream) {
    const float* x    = (const float*)d_in[0];
    const float* W    = (const float*)d_in[1];
    const float* Wr   = (const float*)d_in[2];
    const float* bias = (const float*)d_in[3];
    const int*   esrc = (const int*)d_in[4];
    const int*   edst = (const int*)d_in[5];
    float*       out  = (float*)d_out;

    // ws layout: sum[N_NODES*D] floats, then deg[N_NODES] floats (contiguous)
    float* sum = (float*)d_ws;
    float* deg = sum + (size_t)N_NODES * D;

    const int n4 = (N_NODES * D + N_NODES) / 4;        // 3,225,000 float4s
    sage_zero_kernel<<<(n4 + 255) / 256, 256, 0, stream>>>((float4*)d_ws, n4);

    const int nt = N_EDGES * 32;                       // one wave per edge
    sage_scatter_kernel<<<(nt + 255) / 256, 256, 0, stream>>>(x, esrc, edst, sum, deg);

    sage_invdeg_kernel<<<(N_NODES + 255) / 256, 256, 0, stream>>>(deg);

    const int rowblocks = (N_NODES + 16 * ROWTILES - 1) / (16 * ROWTILES);  // 1563
    sage_gemm_kernel<<<rowblocks, 256, 0, stream>>>(sum, deg, x, W, Wr, bias, out);
}